// HeightChannelAtt_60533269070298
// MI455X (gfx1250) — compile-verified
//
#include <hip/hip_runtime.h>
#include <math.h>

// The Python harness owns main(), I/O, device malloc/memcpy, and timing.

typedef float v2f __attribute__((ext_vector_type(2)));
typedef float v8f __attribute__((ext_vector_type(8)));

// D = A(16x4,f32) * B(4x16,f32) + C(16x16,f32), full-precision CDNA5 WMMA
#define WMMA4(A_, B_, C_) \
  __builtin_amdgcn_wmma_f32_16x16x4_f32(false, (A_), false, (B_), (short)0, (C_), false, false)

__device__ __forceinline__ float rowmax16(float v) {
  v = fmaxf(v, __shfl_xor(v, 1, 32));
  v = fmaxf(v, __shfl_xor(v, 2, 32));
  v = fmaxf(v, __shfl_xor(v, 4, 32));
  v = fmaxf(v, __shfl_xor(v, 8, 32));
  return v;   // uniform across each 16-lane half
}
__device__ __forceinline__ float rowsum16(float v) {
  v += __shfl_xor(v, 1, 32);
  v += __shfl_xor(v, 2, 32);
  v += __shfl_xor(v, 4, 32);
  v += __shfl_xor(v, 8, 32);
  return v;
}

// ---------------------------------------------------------------------------
// Kernel 1: fused QKV projection. One wave per 16-row tile of flattened [B*C,16].
// Q = X @ Wq^T + bq  (torch Linear semantics), likewise K, V. 12 WMMAs / wave.
// ---------------------------------------------------------------------------
__global__ __launch_bounds__(32)
void qkv_proj_kernel(const float* __restrict__ X,
                     const float* __restrict__ Wq, const float* __restrict__ bq,
                     const float* __restrict__ Wk, const float* __restrict__ bk,
                     const float* __restrict__ Wv, const float* __restrict__ bv,
                     float* __restrict__ Qo, float* __restrict__ Ko, float* __restrict__ Vo)
{
  const int  l    = threadIdx.x;
  const int  n    = l & 15;        // A row / B col / C col
  const int  half = l >> 4;
  const long rbase = (long)blockIdx.x * 16;

  // A operand: X rows, A[m,k] pattern -> contiguous f32 pairs from row m=n
  v2f a[4];
#pragma unroll
  for (int p = 0; p < 4; ++p)
    a[p] = *(const v2f*)(X + (rbase + n) * 16 + 4 * p + 2 * half);

  const float* Ws[3] = {Wq, Wk, Wv};
  const float* bs[3] = {bq, bk, bv};
  float*       Os[3] = {Qo, Ko, Vo};

#pragma unroll
  for (int t = 0; t < 3; ++t) {
    // B operand: B[d, e] = W[e][d]; lane's col e = n -> contiguous pairs of row e of W
    v2f wb[4];
#pragma unroll
    for (int p = 0; p < 4; ++p)
      wb[p] = *(const v2f*)(Ws[t] + n * 16 + 4 * p + 2 * half);

    const float bias = bs[t][n];                 // C[m, e] = b[e], e = lane's col
    v8f c = {bias, bias, bias, bias, bias, bias, bias, bias};
#pragma unroll
    for (int p = 0; p < 4; ++p) c = WMMA4(a[p], wb[p], c);

    // C/D layout: vgpr j -> row j + 8*half, col n
#pragma unroll
    for (int j = 0; j < 8; ++j)
      Os[t][(rbase + j + 8 * half) * 16 + n] = c[j];
  }
}

// ---------------------------------------------------------------------------
// Kernel 2: flash-attention over C=1024 keys. One wave per (batch, 16-query tile).
// S = Q K^T (unscaled, as in reference) -> online softmax -> acc += P V.
// P is converted C/D-layout -> A-layout through LDS (stride 18 floats).
// ---------------------------------------------------------------------------
__global__ __launch_bounds__(32)
void attn_kernel(const float* __restrict__ X,
                 const float* __restrict__ Q,
                 const float* __restrict__ K,
                 const float* __restrict__ V,
                 float* __restrict__ out)
{
  __shared__ float Pt[16 * 18];    // 18-float row stride: b64-aligned, conflict-light

  const int  C     = 1024;
  const int  b     = blockIdx.x >> 6;   // C/16 = 64 query tiles per batch
  const int  qt    = blockIdx.x & 63;
  const int  l     = threadIdx.x;
  const int  n     = l & 15;
  const int  half  = l >> 4;
  const long qbase = (long)b * C + qt * 16;

  // Q tile as A operand (resident for the whole loop)
  v2f aq[4];
#pragma unroll
  for (int p = 0; p < 4; ++p)
    aq[p] = *(const v2f*)(Q + (qbase + n) * 16 + 4 * p + 2 * half);

  v8f  acc = {0.f, 0.f, 0.f, 0.f, 0.f, 0.f, 0.f, 0.f};
  float mrun[8], lrun[8];
#pragma unroll
  for (int j = 0; j < 8; ++j) { mrun[j] = -INFINITY; lrun[j] = 0.f; }

  for (int kt = 0; kt < 64; ++kt) {
    const long kbase = (long)b * C + kt * 16;

    if (kt + 1 < 64) {                       // global_prefetch_b8 for next tile
      __builtin_prefetch(K + (kbase + 16) * 16 + l * 8, 0, 3);
      __builtin_prefetch(V + (kbase + 16) * 16 + l * 8, 0, 3);
    }

    // K tile as B operand for S = Q·K^T: B[d, key] = K[key][d]
    v2f bk_[4];
#pragma unroll
    for (int p = 0; p < 4; ++p)
      bk_[p] = *(const v2f*)(K + (kbase + n) * 16 + 4 * p + 2 * half);

    v8f s = {0.f, 0.f, 0.f, 0.f, 0.f, 0.f, 0.f, 0.f};
#pragma unroll
    for (int p = 0; p < 4; ++p) s = WMMA4(aq[p], bk_[p], s);

    // V tile as B operand for acc += P·V: B[k, e] = V[k][e] (column gather per lane)
    v2f bv_[4];
#pragma unroll
    for (int p = 0; p < 4; ++p) {
      const int k0 = 4 * p + 2 * half;
      bv_[p][0] = V[(kbase + k0)     * 16 + n];
      bv_[p][1] = V[(kbase + k0 + 1) * 16 + n];
    }

    __syncthreads();   // previous iteration's LDS reads done before overwrite
#pragma unroll
    for (int j = 0; j < 8; ++j) {
      const float mt   = rowmax16(s[j]);
      const float mnew = fmaxf(mrun[j], mt);
      const float corr = __expf(mrun[j] - mnew);     // exp(-inf)=0 on first tile
      const float pv   = __expf(s[j] - mnew);
      lrun[j] = lrun[j] * corr + rowsum16(pv);
      mrun[j] = mnew;
      acc[j] *= corr;                                 // per-row rescale (uniform/half)
      Pt[(j + 8 * half) * 18 + n] = pv;               // C/D layout -> LDS row-major
    }
    __syncthreads();   // P visible before A-layout reload

#pragma unroll
    for (int p = 0; p < 4; ++p) {
      const v2f ap = *(const v2f*)(Pt + n * 18 + 4 * p + 2 * half);  // A[m=n, k]
      acc = WMMA4(ap, bv_[p], acc);
    }
  }

  // normalize + elementwise input gate
#pragma unroll
  for (int j = 0; j < 8; ++j) {
    const long r = qbase + j + 8 * half;
    out[r * 16 + n] = (acc[j] / lrun[j]) * X[r * 16 + n];
  }
}

extern "C" void kernel_launch(void* const* d_in, const int* in_sizes, int n_in,
                              void* d_out, int out_size, void* d_ws, size_t ws_size,
                              hipStream_t stream) {
  const float* X  = (const float*)d_in[0];
  const float* Wq = (const float*)d_in[1];
  const float* bq = (const float*)d_in[2];
  const float* Wk = (const float*)d_in[3];
  const float* bk = (const float*)d_in[4];
  const float* Wv = (const float*)d_in[5];
  const float* bv = (const float*)d_in[6];
  float* out = (float*)d_out;

  const long BC = 32L * 1024L;                  // flattened rows
  float* Qw = (float*)d_ws;                     // 3 x 2 MB workspace
  float* Kw = Qw + BC * 16;
  float* Vw = Kw + BC * 16;

  const int tiles = (int)(BC / 16);             // 2048 waves each
  qkv_proj_kernel<<<tiles, 32, 0, stream>>>(X, Wq, bq, Wk, bk, Wv, bv, Qw, Kw, Vw);
  attn_kernel<<<tiles, 32, 0, stream>>>(X, Qw, Kw, Vw, out);
}